// GCNLayer_816043786790
// MI455X (gfx1250) — compile-verified
//
#include <hip/hip_runtime.h>
#include <hip/hip_bf16.h>

typedef __attribute__((ext_vector_type(2))) float v2f;
typedef __attribute__((ext_vector_type(8))) float v8f;

#define D_IN   128
#define D_OUT  128
#define LDA    132   // padded LDS stride for A tile (kills bank conflicts on column reads)
#define LDW    132   // padded LDS stride for W

// ---------------------------------------------------------------------------
// zero fill (float4 vectorized, grid-stride)
// ---------------------------------------------------------------------------
__global__ void gcn_zero_f4(float4* __restrict__ p, long long n4) {
    long long i = (long long)blockIdx.x * blockDim.x + threadIdx.x;
    long long stride = (long long)gridDim.x * blockDim.x;
    float4 z = make_float4(0.f, 0.f, 0.f, 0.f);
    for (; i < n4; i += stride) p[i] = z;
}

// ---------------------------------------------------------------------------
// degree counting: out_deg[src[e]]++, in_deg[dst[e]]++
// ---------------------------------------------------------------------------
__global__ void gcn_degree(const int* __restrict__ src, const int* __restrict__ dst,
                           unsigned* __restrict__ odeg, unsigned* __restrict__ ideg, int E) {
    int e = blockIdx.x * blockDim.x + threadIdx.x;
    if (e < E) {
        atomicAdd(&odeg[src[e]], 1u);
        atomicAdd(&ideg[dst[e]], 1u);
    }
}

// ---------------------------------------------------------------------------
// node_f = concat(u_f @ u_w, v_f @ v_w) * rsqrt(max(out_deg,1))
// One 16-row tile per block; 8 waves, each owns one 16-wide N tile.
// K loop: 32 x V_WMMA_F32_16X16X4_F32 (fp32 in/out).
// ---------------------------------------------------------------------------
__global__ __launch_bounds__(256) void gcn_gemm_wmma(
    const float* __restrict__ u_f, const float* __restrict__ v_f,
    const float* __restrict__ u_w, const float* __restrict__ v_w,
    const unsigned* __restrict__ odeg, float* __restrict__ node_f,
    int NU, int NV, int tilesU) {

    extern __shared__ float smem[];
    float* As = smem;                 // 16 x LDA
    float* Ws = smem + 16 * LDA;      // 128 x LDW

    const int t   = blockIdx.x;
    const int tid = threadIdx.x;

    const bool isU       = (t < tilesU);
    const int  baseLocal = (isU ? t : (t - tilesU)) * 16;
    const int  rowsPart  = isU ? NU : NV;
    const int  globalOff = isU ? 0 : NU;
    const float* X = (isU ? u_f : v_f) + (size_t)baseLocal * D_IN;
    const float* W =  isU ? u_w : v_w;

    // stage A tile: 16 x 128 f32 (2 float4 per thread)
    for (int i = tid; i < (16 * D_IN) / 4; i += 256) {
        int r = i >> 5;               // i*4 / 128
        int c = (i & 31) << 2;        // i*4 % 128
        float4 v = make_float4(0.f, 0.f, 0.f, 0.f);
        if (baseLocal + r < rowsPart) v = *(const float4*)(X + (size_t)r * D_IN + c);
        float* a = &As[r * LDA + c];
        a[0] = v.x; a[1] = v.y; a[2] = v.z; a[3] = v.w;
    }
    // stage full W: 128 x 128 f32 (16 float4 per thread)
    for (int i = tid; i < (D_IN * D_OUT) / 4; i += 256) {
        int r = i >> 5;
        int c = (i & 31) << 2;
        float4 v = *(const float4*)(W + (size_t)r * D_OUT + c);
        float* w = &Ws[r * LDW + c];
        w[0] = v.x; w[1] = v.y; w[2] = v.z; w[3] = v.w;
    }
    __syncthreads();

    const int wave = tid >> 5;        // wave32
    const int lane = tid & 31;
    const int n0   = wave * 16;       // this wave's N tile
    const int mn   = lane & 15;       // row M (for A) / col N (for B)
    const int kk   = (lane >> 4) * 2; // K sub-offset: lanes 0-15 -> {0,1}, 16-31 -> {2,3}

    v8f acc = {};
    #pragma unroll
    for (int k0 = 0; k0 < D_IN; k0 += 4) {
        v2f a, b;
        a.x = As[mn * LDA + k0 + kk];
        a.y = As[mn * LDA + k0 + kk + 1];
        b.x = Ws[(k0 + kk) * LDW + n0 + mn];
        b.y = Ws[(k0 + kk + 1) * LDW + n0 + mn];
        acc = __builtin_amdgcn_wmma_f32_16x16x4_f32(
            /*neg_a=*/false, a, /*neg_b=*/false, b,
            /*c_mod=*/(short)0, acc, /*reuse_a=*/false, /*reuse_b=*/false);
    }

    // epilogue: scale by rsqrt(max(out_deg,1)) and store
    const int mhi = (lane >> 4) * 8;  // C layout: VGPR r holds M = r + 8*(lane>=16)
    #pragma unroll
    for (int r = 0; r < 8; ++r) {
        int m = r + mhi;
        if (baseLocal + m < rowsPart) {
            size_t row = (size_t)(globalOff + baseLocal + m);
            float s = rsqrtf(fmaxf((float)odeg[row], 1.0f));
            node_f[row * D_OUT + n0 + mn] = acc[r] * s;
        }
    }
}

// ---------------------------------------------------------------------------
// edge phase: out[dst[e]] += node_f[src[e]] + e_f[e]
// one wave per edge; lane moves a float4; native f32 atomics (L2-resident out)
// ---------------------------------------------------------------------------
__global__ __launch_bounds__(256) void gcn_edge_scatter(
    const float* __restrict__ node_f, const float* __restrict__ e_f,
    const int* __restrict__ src, const int* __restrict__ dst,
    float* __restrict__ out, int E) {

    int e = blockIdx.x * 8 + (threadIdx.x >> 5);  // wave-uniform
    if (e >= E) return;
    int lane = threadIdx.x & 31;
    int s = src[e];
    int d = dst[e];

    const float4 nf = *(const float4*)(node_f + (size_t)s * D_OUT + lane * 4);
    const float4 ef = *(const float4*)(e_f    + (size_t)e * D_OUT + lane * 4);
    float* o = out + (size_t)d * D_OUT + lane * 4;
    unsafeAtomicAdd(o + 0, nf.x + ef.x);
    unsafeAtomicAdd(o + 1, nf.y + ef.y);
    unsafeAtomicAdd(o + 2, nf.z + ef.z);
    unsafeAtomicAdd(o + 3, nf.w + ef.w);
}

// ---------------------------------------------------------------------------
// final: out[i,:] *= rsqrt(max(in_deg[i],1)); float4 per thread
// ---------------------------------------------------------------------------
__global__ void gcn_norm(float* __restrict__ out, const unsigned* __restrict__ ideg, int N) {
    int i = blockIdx.x * blockDim.x + threadIdx.x;  // N * 32 work items
    int row = i >> 5;
    if (row >= N) return;
    int c = (i & 31) * 4;
    float s = rsqrtf(fmaxf((float)ideg[row], 1.0f));
    float4* p = (float4*)(out + (size_t)row * D_OUT + c);
    float4 v = *p;
    v.x *= s; v.y *= s; v.z *= s; v.w *= s;
    *p = v;
}

// ---------------------------------------------------------------------------
extern "C" void kernel_launch(void* const* d_in, const int* in_sizes, int n_in,
                              void* d_out, int out_size, void* d_ws, size_t ws_size,
                              hipStream_t stream) {
    const float* u_f = (const float*)d_in[0];
    const float* v_f = (const float*)d_in[1];
    const float* e_f = (const float*)d_in[2];
    const float* u_w = (const float*)d_in[3];
    const float* v_w = (const float*)d_in[4];
    const int*   src = (const int*)d_in[5];
    const int*   dst = (const int*)d_in[6];
    float* out = (float*)d_out;

    const int NU = in_sizes[0] / D_IN;
    const int NV = in_sizes[1] / D_IN;
    const int E  = in_sizes[5];
    const int N  = NU + NV;

    // workspace layout
    float*    node_f = (float*)d_ws;                       // N * 128 f32
    unsigned* odeg   = (unsigned*)(node_f + (size_t)N * D_OUT);
    unsigned* ideg   = odeg + N;

    // 1) zero output + degree counters
    {
        long long n4 = (long long)N * D_OUT / 4;
        int blocks = (int)((n4 + 255) / 256);
        if (blocks > 65535) blocks = 65535;
        gcn_zero_f4<<<blocks, 256, 0, stream>>>((float4*)out, n4);
        long long d4 = (2LL * N + 3) / 4;
        gcn_zero_f4<<<(int)((d4 + 255) / 256), 256, 0, stream>>>((float4*)odeg, d4);
    }

    // 2) degrees
    gcn_degree<<<(E + 255) / 256, 256, 0, stream>>>(src, dst, odeg, ideg, E);

    // 3) projection GEMMs (WMMA fp32) fused with out-degree normalization
    {
        int tilesU = (NU + 15) / 16;
        int tilesV = (NV + 15) / 16;
        size_t shmem = (size_t)(16 * LDA + D_IN * LDW) * sizeof(float);  // ~76 KB
        gcn_gemm_wmma<<<tilesU + tilesV, 256, shmem, stream>>>(
            u_f, v_f, u_w, v_w, odeg, node_f, NU, NV, tilesU);
    }

    // 4) gather + edge add + scatter-sum (one wave per edge)
    gcn_edge_scatter<<<(E + 7) / 8, 256, 0, stream>>>(node_f, e_f, src, dst, out, E);

    // 5) in-degree normalization
    {
        long long work = (long long)N * 32;
        gcn_norm<<<(int)((work + 255) / 256), 256, 0, stream>>>(out, ideg, N);
    }
}